// SAGEClassic_11390253269762
// MI455X (gfx1250) — compile-verified
//
#include <hip/hip_runtime.h>
#include <math.h>

typedef float v2f __attribute__((ext_vector_type(2)));
typedef float v8f __attribute__((ext_vector_type(8)));

// ---------------- fill ----------------
__global__ void fill0_kernel(float* __restrict__ p, long long n) {
    long long i = (long long)blockIdx.x * blockDim.x + threadIdx.x;
    if (i < n) p[i] = 0.0f;
}

// ---------------- scatter (segment-sum) ----------------
__global__ void scatter_feat_kernel(const float* __restrict__ x,
                                    const int* __restrict__ src,
                                    const int* __restrict__ dst,
                                    float* __restrict__ agg,
                                    long long total, int F) {
    long long idx = (long long)blockIdx.x * blockDim.x + threadIdx.x;
    if (idx >= total) return;
    int e = (int)(idx / F);
    int f = (int)(idx % F);
    int s = src[e];
    int d = dst[e];
    float v = x[(long long)s * F + f];
    __hip_atomic_fetch_add(&agg[(long long)d * F + f], v,
                           __ATOMIC_RELAXED, __HIP_MEMORY_SCOPE_AGENT);
}

__global__ void scatter_cnt_kernel(const int* __restrict__ dst,
                                   float* __restrict__ cnt, int E) {
    int e = blockIdx.x * blockDim.x + threadIdx.x;
    if (e >= E) return;
    __hip_atomic_fetch_add(&cnt[dst[e]], 1.0f,
                           __ATOMIC_RELAXED, __HIP_MEMORY_SCOPE_AGENT);
}

// ---------------- mean = agg / max(cnt,1) (in place) ----------------
__global__ void finalize_mean_kernel(float* __restrict__ agg,
                                     const float* __restrict__ cnt,
                                     long long total, int F) {
    long long idx = (long long)blockIdx.x * blockDim.x + threadIdx.x;
    if (idx >= total) return;
    int n = (int)(idx / F);
    float c = fmaxf(cnt[n], 1.0f);
    agg[idx] = agg[idx] / c;
}

// ---------------- fused dual GEMM: out = A1@B1 + A2@B2 + bias (opt relu) ----
// One wave32 per 16x16 output tile, fp32 WMMA 16x16x4.
// A1 = mean (N x K), A2 = x_tgt (N x K), B1 = Wl (K x D), B2 = Wr (K x D).
// N % 16 == 0, K % 4 == 0. D handled with column masking (Dtiles = ceil(D/16)).
__global__ void gemm_dual_wmma_kernel(const float* __restrict__ A1,
                                      const float* __restrict__ A2,
                                      const float* __restrict__ B1,
                                      const float* __restrict__ B2,
                                      const float* __restrict__ bias,
                                      float* __restrict__ out,
                                      int N, int K, int D, int Dtiles, int relu) {
    int wavesPerBlock = blockDim.x >> 5;
    int wave = blockIdx.x * wavesPerBlock + (threadIdx.x >> 5);
    int totalTiles = (N >> 4) * Dtiles;
    if (wave >= totalTiles) return;          // wave-uniform: EXEC stays all-ones

    int tm = wave / Dtiles;
    int tn = wave % Dtiles;

    int lane = threadIdx.x & 31;
    int half = lane >> 4;                    // 0: K pair {0,1}; 1: K pair {2,3}
    int lid  = lane & 15;

    int arow = (tm << 4) + lid;              // A row for this lane (always < N)
    int col  = (tn << 4) + lid;              // B/C column for this lane
    bool colOK = col < D;
    int colc  = colOK ? col : (D - 1);       // clamped, always in-bounds
    float csel = colOK ? 1.0f : 0.0f;

    const float* a1p = A1 + (long long)arow * K;
    const float* a2p = A2 + (long long)arow * K;

    v8f acc = {};
    for (int k0 = 0; k0 < K; k0 += 4) {
        int k = k0 + (half << 1);

        v2f a1; a1.x = a1p[k]; a1.y = a1p[k + 1];
        v2f b1; b1.x = B1[(long long)k * D + colc] * csel;
                b1.y = B1[(long long)(k + 1) * D + colc] * csel;
        acc = __builtin_amdgcn_wmma_f32_16x16x4_f32(
                  false, a1, false, b1, (short)0, acc, false, false);

        v2f a2; a2.x = a2p[k]; a2.y = a2p[k + 1];
        v2f b2; b2.x = B2[(long long)k * D + colc] * csel;
                b2.y = B2[(long long)(k + 1) * D + colc] * csel;
        acc = __builtin_amdgcn_wmma_f32_16x16x4_f32(
                  false, a2, false, b2, (short)0, acc, false, false);
    }

    float bv = bias[colc];
    int rbase = (tm << 4) + (half << 3);     // C/D: vgpr r -> row r (+8 for hi lanes)
    if (colOK) {
#pragma unroll
        for (int r = 0; r < 8; ++r) {
            float v = acc[r] + bv;
            if (relu) v = fmaxf(v, 0.0f);
            out[(long long)(rbase + r) * D + col] = v;
        }
    }
}

// ---------------- log_softmax over rows ----------------
__global__ void log_softmax_kernel(const float* __restrict__ in,
                                   float* __restrict__ out, int R, int C) {
    int r = blockIdx.x * blockDim.x + threadIdx.x;
    if (r >= R) return;
    const float* p = in + (long long)r * C;
    float m = -INFINITY;
    for (int j = 0; j < C; ++j) m = fmaxf(m, p[j]);
    float s = 0.0f;
    for (int j = 0; j < C; ++j) s += expf(p[j] - m);
    float l = logf(s);
    float* o = out + (long long)r * C;
    for (int j = 0; j < C; ++j) o[j] = p[j] - m - l;
}

// ---------------- host-side orchestration ----------------
static inline unsigned cdivu(long long a, long long b) {
    return (unsigned)((a + b - 1) / b);
}

extern "C" void kernel_launch(void* const* d_in, const int* in_sizes, int n_in,
                              void* d_out, int out_size, void* d_ws, size_t ws_size,
                              hipStream_t stream) {
    const float* x    = (const float*)d_in[0];    // 900000 x 100
    const int*   src0 = (const int*)d_in[1];      // 900000
    const int*   dst0 = (const int*)d_in[2];
    const int*   src1 = (const int*)d_in[3];      // 60000
    const int*   dst1 = (const int*)d_in[4];
    const int*   src2 = (const int*)d_in[5];      // 5120
    const int*   dst2 = (const int*)d_in[6];
    const float* Wl0  = (const float*)d_in[7];    // 100 x 256
    const float* Wr0  = (const float*)d_in[8];
    const float* b0   = (const float*)d_in[9];
    const float* Wl1  = (const float*)d_in[10];   // 256 x 256
    const float* Wr1  = (const float*)d_in[11];
    const float* b1   = (const float*)d_in[12];
    const float* Wl2  = (const float*)d_in[13];   // 256 x 47
    const float* Wr2  = (const float*)d_in[14];
    const float* b2   = (const float*)d_in[15];

    // Workspace carve-out (floats)
    float* ws   = (float*)d_ws;
    float* agg0 = ws;                         // 60000*100   = 6,000,000
    float* cnt0 = agg0 + 6000000LL;           // 60,000
    float* h1   = cnt0 + 60000LL;             // 60000*256   = 15,360,000
    float* agg1 = h1   + 15360000LL;          // 6000*256    = 1,536,000
    float* cnt1 = agg1 + 1536000LL;           // 6,000
    float* h2   = cnt1 + 6000LL;              // 6000*256    = 1,536,000
    float* agg2 = h2   + 1536000LL;           // 1024*256    = 262,144
    float* cnt2 = agg2 + 262144LL;            // 1,024
    float* lg   = cnt2 + 1024LL;              // 1024*47     = 48,128

    const int T = 256;  // 8 waves per block

    // ----- Layer 0: E=900000, F=100, Ntgt=60000, K=100, D=256, relu -----
    fill0_kernel<<<cdivu(6060000, T), T, 0, stream>>>(agg0, 6060000LL); // agg0|cnt0
    scatter_feat_kernel<<<cdivu(90000000LL, T), T, 0, stream>>>(
        x, src0, dst0, agg0, 90000000LL, 100);
    scatter_cnt_kernel<<<cdivu(900000, T), T, 0, stream>>>(dst0, cnt0, 900000);
    finalize_mean_kernel<<<cdivu(6000000LL, T), T, 0, stream>>>(
        agg0, cnt0, 6000000LL, 100);
    {
        int tiles = (60000 / 16) * 16;  // 60000 tiles (Dtiles=16)
        gemm_dual_wmma_kernel<<<cdivu(tiles, 8), T, 0, stream>>>(
            agg0, x, Wl0, Wr0, b0, h1, 60000, 100, 256, 16, 1);
    }

    // ----- Layer 1: E=60000, F=256, Ntgt=6000, K=256, D=256, relu -----
    fill0_kernel<<<cdivu(1542000, T), T, 0, stream>>>(agg1, 1542000LL); // agg1|cnt1
    scatter_feat_kernel<<<cdivu(15360000LL, T), T, 0, stream>>>(
        h1, src1, dst1, agg1, 15360000LL, 256);
    scatter_cnt_kernel<<<cdivu(60000, T), T, 0, stream>>>(dst1, cnt1, 60000);
    finalize_mean_kernel<<<cdivu(1536000LL, T), T, 0, stream>>>(
        agg1, cnt1, 1536000LL, 256);
    {
        int tiles = (6000 / 16) * 16;   // 6000 tiles
        gemm_dual_wmma_kernel<<<cdivu(tiles, 8), T, 0, stream>>>(
            agg1, h1, Wl1, Wr1, b1, h2, 6000, 256, 256, 16, 1);
    }

    // ----- Layer 2: E=5120, F=256, Ntgt=1024, K=256, D=47, no relu -----
    fill0_kernel<<<cdivu(263168, T), T, 0, stream>>>(agg2, 263168LL);   // agg2|cnt2
    scatter_feat_kernel<<<cdivu(1310720LL, T), T, 0, stream>>>(
        h2, src2, dst2, agg2, 1310720LL, 256);
    scatter_cnt_kernel<<<cdivu(5120, T), T, 0, stream>>>(dst2, cnt2, 5120);
    finalize_mean_kernel<<<cdivu(262144LL, T), T, 0, stream>>>(
        agg2, cnt2, 262144LL, 256);
    {
        int tiles = (1024 / 16) * 3;    // Dtiles = ceil(47/16) = 3 -> 192 tiles
        gemm_dual_wmma_kernel<<<cdivu(tiles, 8), T, 0, stream>>>(
            agg2, h2, Wl2, Wr2, b2, lg, 1024, 256, 47, 3, 0);
    }

    // ----- log_softmax (1024 x 47) -> d_out -----
    log_softmax_kernel<<<cdivu(1024, T), T, 0, stream>>>(lg, (float*)d_out, 1024, 47);
}